// EncoderLayer_20263655702649
// MI455X (gfx1250) — compile-verified
//
#include <hip/hip_runtime.h>
#include <hip/hip_bf16.h>

typedef __attribute__((ext_vector_type(2))) float v2f;
typedef __attribute__((ext_vector_type(8))) float v8f;
typedef int v4i __attribute__((vector_size(16)));   // pointee type for async-LDS builtin

#define D_FEAT 128
#define BN_EPS 1e-5f

#if defined(__has_builtin)
#if __has_builtin(__builtin_amdgcn_global_load_async_to_lds_b128)
#define HAS_ASYNC_LDS 1
#endif
#endif

// ---------------------------------------------------------------------------
// init: deg = 1 (self loop), zero the 4 column-stat accumulators
// ---------------------------------------------------------------------------
__global__ __launch_bounds__(256) void k_init(float* __restrict__ deg,
                                              float* __restrict__ s1, float* __restrict__ q1,
                                              float* __restrict__ s2, float* __restrict__ q2,
                                              int n) {
  int i = blockIdx.x * 256 + threadIdx.x;
  if (i < n) deg[i] = 1.0f;
  if (i < D_FEAT) { s1[i] = 0.f; q1[i] = 0.f; s2[i] = 0.f; q2[i] = 0.f; }
}

// ---------------------------------------------------------------------------
// WMMA GEMM:  C[nrows x 128] = op(A)[nrows x 128] * W[128 x 128]
//   TRANS: op(A) = relu(A*tsc[c] + tsh[c])  (fused BN+ReLU on input)
//   STATS: accumulate per-column sum / sum-of-squares of C via atomics.
// 8 waves / 256 threads; wave w owns column panel n0 = 16*w; 4 row tiles / WG.
// A tile staged via GLOBAL_LOAD_ASYNC_TO_LDS_B128 when no transform is needed.
// ---------------------------------------------------------------------------
template <bool TRANS, bool STATS>
__global__ __launch_bounds__(256) void k_gemm_wmma(const float* __restrict__ A,
                                                   const float* __restrict__ W,
                                                   float* __restrict__ C, int nrows,
                                                   const float* __restrict__ tsc,
                                                   const float* __restrict__ tsh,
                                                   float* __restrict__ colsum,
                                                   float* __restrict__ colsq) {
  __shared__ float At[64][132];  // pad 128->132: A-frag reads hit distinct banks

  const int tid  = threadIdx.x;
  const int wave = tid >> 5;
  const int lane = tid & 31;
  const int lr   = lane & 15;     // row within A-frag / col within B-frag
  const int hi   = lane >> 4;     // K-half selector
  const int n0   = wave * 16;
  const long rowBase = (long)blockIdx.x * 64;

  // ---- stage 64x128 A tile into LDS ----------------------------------------
#if defined(HAS_ASYNC_LDS)
  if constexpr (!TRANS) {
    // async DMA path: memory -> LDS without touching VGPRs (ASYNCcnt tracked)
    for (int i = 0; i < 8; ++i) {
      int idx = tid + i * 256;    // float4 slot in 64 x 32
      int r   = idx >> 5;
      int c4  = (idx & 31) << 2;
      long gr = rowBase + r;
      if (gr < nrows) {
        __builtin_amdgcn_global_load_async_to_lds_b128(
            (__attribute__((address_space(1))) v4i*)(A + gr * D_FEAT + c4),
            (__attribute__((address_space(3))) v4i*)&At[r][c4], 0, 0);
      } else {
        *(float4*)&At[r][c4] = make_float4(0.f, 0.f, 0.f, 0.f);
      }
    }
    asm volatile("s_wait_asynccnt 0x0" ::: "memory");
  } else
#endif
  {
    for (int i = 0; i < 8; ++i) {
      int idx = tid + i * 256;    // float4 slot in 64 x 32
      int r   = idx >> 5;
      int c4  = (idx & 31) << 2;
      long gr = rowBase + r;
      float4 v = make_float4(0.f, 0.f, 0.f, 0.f);
      if (gr < nrows) v = *(const float4*)(A + gr * D_FEAT + c4);
      if constexpr (TRANS) {  // fused BatchNorm + ReLU on the input of GEMM2
        v.x = fmaxf(v.x * tsc[c4 + 0] + tsh[c4 + 0], 0.f);
        v.y = fmaxf(v.y * tsc[c4 + 1] + tsh[c4 + 1], 0.f);
        v.z = fmaxf(v.z * tsc[c4 + 2] + tsh[c4 + 2], 0.f);
        v.w = fmaxf(v.w * tsc[c4 + 3] + tsh[c4 + 3], 0.f);
      }
      *(float4*)&At[r][c4] = v;
    }
  }
  __syncthreads();

  // ---- preload full 128x16 B panel into registers (reused for 4 row tiles) -
  v2f bp[32];
#pragma unroll
  for (int k = 0; k < 32; ++k) {
    int kk = 4 * k + 2 * hi;
    bp[k].x = W[(size_t)kk * D_FEAT + n0 + lr];
    bp[k].y = W[(size_t)(kk + 1) * D_FEAT + n0 + lr];
  }

  float ssum = 0.f, ssq = 0.f;
  for (int rt = 0; rt < 4; ++rt) {
    long rb = rowBase + rt * 16;
    if (rb >= nrows) break;       // wave-uniform; EXEC stays all-ones for WMMA

    v8f acc0 = {}, acc1 = {};
#pragma unroll
    for (int k = 0; k < 32; k += 2) {
      v2f a0, a1;
      int kk0 = 4 * k + 2 * hi;
      int kk1 = kk0 + 4;
      a0.x = At[rt * 16 + lr][kk0];
      a0.y = At[rt * 16 + lr][kk0 + 1];
      a1.x = At[rt * 16 + lr][kk1];
      a1.y = At[rt * 16 + lr][kk1 + 1];
      acc0 = __builtin_amdgcn_wmma_f32_16x16x4_f32(false, a0, false, bp[k],
                                                   (short)0, acc0, false, false);
      acc1 = __builtin_amdgcn_wmma_f32_16x16x4_f32(false, a1, false, bp[k + 1],
                                                   (short)0, acc1, false, false);
    }
#pragma unroll
    for (int v = 0; v < 8; ++v) {
      float val = acc0[v] + acc1[v];
      C[(size_t)(rb + v + 8 * hi) * D_FEAT + n0 + lr] = val;
      if constexpr (STATS) {
        ssum += val;
        ssq  += val * val;
      }
    }
  }
  if constexpr (STATS) {
    atomicAdd(&colsum[n0 + lr], ssum);
    atomicAdd(&colsq[n0 + lr], ssq);
  }
}

// ---------------------------------------------------------------------------
// fold column sums into BN scale/shift:  y = x*a[c] + b[c]
// ---------------------------------------------------------------------------
__global__ void k_bn_prep(const float* __restrict__ colsum, const float* __restrict__ colsq,
                          const float* __restrict__ gamma, const float* __restrict__ beta,
                          float* __restrict__ a, float* __restrict__ b, float invN) {
  int c = threadIdx.x;
  float mean = colsum[c] * invN;
  float var  = colsq[c] * invN - mean * mean;
  float sc   = gamma[c] * rsqrtf(var + BN_EPS);
  a[c] = sc;
  b[c] = beta[c] - mean * sc;
}

// ---------------------------------------------------------------------------
// edge degree count at dst
// ---------------------------------------------------------------------------
__global__ __launch_bounds__(256) void k_deg(const int* __restrict__ dst,
                                             float* __restrict__ deg, int E) {
  int e = blockIdx.x * 256 + threadIdx.x;
  if (e < E) atomicAdd(&deg[dst[e]], 1.0f);
}

// deg -> deg^{-1/2} in place (deg >= 1 always, self loop)
__global__ __launch_bounds__(256) void k_dinv(float* __restrict__ deg, int n) {
  int i = blockIdx.x * 256 + threadIdx.x;
  if (i < n) deg[i] = rsqrtf(deg[i]);
}

// self-loop contribution: out = dinv[i]^2 * h2[i]   (plain store, no zero-init)
__global__ __launch_bounds__(256) void k_selfloop(const float* __restrict__ h2,
                                                  const float* __restrict__ dinv,
                                                  float* __restrict__ out, int n) {
  long i = (long)blockIdx.x * 256 + threadIdx.x;       // float4 index
  long node = (i * 4) >> 7;
  if (node >= n) return;
  float w = dinv[node];
  w = w * w;
  float4 v = ((const float4*)h2)[i];
  v.x *= w; v.y *= w; v.z *= w; v.w *= w;
  ((float4*)out)[i] = v;
}

// one wave per edge: out[dst] += dinv[s]*dinv[d] * h2[src]
__global__ __launch_bounds__(256) void k_scatter(const int* __restrict__ src,
                                                 const int* __restrict__ dst,
                                                 const float* __restrict__ dinv,
                                                 const float* __restrict__ h2,
                                                 float* __restrict__ out, int E) {
  long t = (long)blockIdx.x * 256 + threadIdx.x;
  int e = (int)(t >> 5);
  int l = (int)(t & 31);
  if (e >= E) return;
  int s = src[e];
  int d = dst[e];
  const float* hrow = h2 + (size_t)s * D_FEAT;
  __builtin_prefetch(hrow, 0, 0);                       // global_prefetch_b8
  float w = dinv[s] * dinv[d];
  float4 v = ((const float4*)hrow)[l];
  float* orow = out + (size_t)d * D_FEAT + 4 * l;
  atomicAdd(orow + 0, w * v.x);
  atomicAdd(orow + 1, w * v.y);
  atomicAdd(orow + 2, w * v.z);
  atomicAdd(orow + 3, w * v.w);
}

// column stats of X (nrows x 128): 128 threads, 256-row strip per block
__global__ __launch_bounds__(128) void k_colstats(const float* __restrict__ X, int nrows,
                                                  float* __restrict__ colsum,
                                                  float* __restrict__ colsq) {
  int c = threadIdx.x;
  long r0 = (long)blockIdx.x * 256;
  long r1 = r0 + 256;
  if (r1 > nrows) r1 = nrows;
  float s = 0.f, q = 0.f;
  for (long r = r0; r < r1; ++r) {
    float v = X[r * D_FEAT + c];
    s += v;
    q += v * v;
  }
  atomicAdd(&colsum[c], s);
  atomicAdd(&colsq[c], q);
}

// final fused BN + ReLU
__global__ __launch_bounds__(256) void k_bn_relu(const float* __restrict__ X,
                                                 const float* __restrict__ a,
                                                 const float* __restrict__ b,
                                                 float* __restrict__ out, int n) {
  long i = (long)blockIdx.x * 256 + threadIdx.x;        // float4 index
  long node = (i * 4) >> 7;
  if (node >= n) return;
  int c = (int)((i * 4) & (D_FEAT - 1));
  float4 v = ((const float4*)X)[i];
  v.x = fmaxf(v.x * a[c + 0] + b[c + 0], 0.f);
  v.y = fmaxf(v.y * a[c + 1] + b[c + 1], 0.f);
  v.z = fmaxf(v.z * a[c + 2] + b[c + 2], 0.f);
  v.w = fmaxf(v.w * a[c + 3] + b[c + 3], 0.f);
  ((float4*)out)[i] = v;
}

// ---------------------------------------------------------------------------
extern "C" void kernel_launch(void* const* d_in, const int* in_sizes, int n_in,
                              void* d_out, int out_size, void* d_ws, size_t ws_size,
                              hipStream_t stream) {
  const float* x      = (const float*)d_in[0];
  const int*   ei     = (const int*)d_in[1];
  const float* W_mlp  = (const float*)d_in[2];
  // d_in[3] = b_mlp : constant per column before BN1 -> cancelled by mean subtraction
  const float* gamma1 = (const float*)d_in[4];
  const float* beta1  = (const float*)d_in[5];
  const float* W_conv = (const float*)d_in[6];
  // d_in[7] = b_conv : constant per column before BN2 -> cancelled by mean subtraction
  const float* gamma2 = (const float*)d_in[8];
  const float* beta2  = (const float*)d_in[9];

  const int N = in_sizes[0] / D_FEAT;
  const int E = in_sizes[1] / 2;
  const int* src = ei;
  const int* dst = ei + E;

  float* ws  = (float*)d_ws;
  float* h1  = ws;                              // N*128 : x @ W_mlp
  float* h2  = h1 + (size_t)N * D_FEAT;         // N*128 : bn_relu(h1) @ W_conv
  float* op  = h2 + (size_t)N * D_FEAT;         // N*128 : aggregated (pre-BN2)
  float* deg = op + (size_t)N * D_FEAT;         // N     : degree -> dinv in place
  float* s1  = deg + N;
  float* q1  = s1 + D_FEAT;
  float* a1  = q1 + D_FEAT;
  float* b1  = a1 + D_FEAT;
  float* s2  = b1 + D_FEAT;
  float* q2  = s2 + D_FEAT;
  float* a2  = q2 + D_FEAT;
  float* b2  = a2 + D_FEAT;

  const float invN = 1.0f / (float)N;
  const int gN    = (N + 255) / 256;
  const int gGemm = (N + 63) / 64;
  const int gVec4 = (N * (D_FEAT / 4) + 255) / 256;

  // 1) init deg=1, zero stats
  k_init<<<gN, 256, 0, stream>>>(deg, s1, q1, s2, q2, N);
  // 2) GEMM1 + BN1 stats (bias skipped: cancelled by BN); async-LDS staging
  k_gemm_wmma<false, true><<<gGemm, 256, 0, stream>>>(x, W_mlp, h1, N,
                                                      nullptr, nullptr, s1, q1);
  // 3) degree counting (independent of GEMM, same stream)
  k_deg<<<(E + 255) / 256, 256, 0, stream>>>(dst, deg, E);
  // 4) BN1 scale/shift
  k_bn_prep<<<1, D_FEAT, 0, stream>>>(s1, q1, gamma1, beta1, a1, b1, invN);
  // 5) GEMM2 with fused BN1+ReLU on input (manual staging, transform in-flight)
  k_gemm_wmma<true, false><<<gGemm, 256, 0, stream>>>(h1, W_conv, h2, N,
                                                      a1, b1, nullptr, nullptr);
  // 6) deg -> deg^{-1/2}
  k_dinv<<<gN, 256, 0, stream>>>(deg, N);
  // 7) self-loop term initializes out
  k_selfloop<<<gVec4, 256, 0, stream>>>(h2, deg, op, N);
  // 8) edge scatter (1 wave per edge)
  k_scatter<<<(E * 32 + 255) / 256, 256, 0, stream>>>(src, dst, deg, h2, op, E);
  // 9) BN2 stats
  k_colstats<<<(N + 255) / 256, D_FEAT, 0, stream>>>(op, N, s2, q2);
  // 10) BN2 scale/shift
  k_bn_prep<<<1, D_FEAT, 0, stream>>>(s2, q2, gamma2, beta2, a2, b2, invN);
  // 11) final BN + ReLU -> d_out
  k_bn_relu<<<gVec4, 256, 0, stream>>>(op, a2, b2, (float*)d_out, N);
}